// SpectralConvWavelet3D_41455024341105
// MI455X (gfx1250) — compile-verified
//
#include <hip/hip_runtime.h>

// ---------------------------------------------------------------------------
// db4 filters. Analysis correlation kernels are DEC_*[::-1] (use c_*[7-k]);
// synthesis correlation kernels are exactly DEC_* (use c_*[k]), since
// KS = REC[::-1] = DEC.
// ---------------------------------------------------------------------------
__constant__ float c_lo[8] = {
  -0.010597401784997278f,  0.032883011666982945f,
   0.030841381835986965f, -0.18703481171888114f,
  -0.02798376941698385f,   0.6308807679295904f,
   0.7148465705525415f,    0.23037781330885523f };
__constant__ float c_hi[8] = {
  -0.23037781330885523f,   0.7148465705525415f,
  -0.6308807679295904f,   -0.02798376941698385f,
   0.18703481171888114f,   0.030841381835986965f,
  -0.032883011666982945f, -0.010597401784997278f };

typedef __attribute__((ext_vector_type(2))) float v2f;
typedef __attribute__((ext_vector_type(8))) float v8f;

// ---------------------------------------------------------------------------
// Forward analysis pass: stride-2 correlation with reversed filter, periodic
// pad (6, 6+Lin%2). Array viewed as [outer][Lin][inner] -> [outer][Lout][inner]
// (lo block; hi block is a separate, identically-strided pointer).
// ---------------------------------------------------------------------------
template<bool HI>
__global__ __launch_bounds__(256)
void fwd_pass(const float* __restrict__ in, float* __restrict__ outLo,
              float* __restrict__ outHi,
              int Lin, int Lout, long inner, long total)
{
  long idx = (long)blockIdx.x * blockDim.x + threadIdx.x;
  if (idx >= total) return;
  long ii = idx % inner;
  long t  = idx / inner;
  int  j  = (int)(t % Lout);
  long oa = t / Lout;

  const float* src = in + oa * (long)Lin * inner + ii;
  float lo = 0.f, hi = 0.f;
  int base = 2 * j - 6;
#pragma unroll
  for (int k = 0; k < 8; ++k) {
    int z = base + k;                      // range [-6, 2*Lout+1) — single wrap
    if (z < 0) z += Lin; else if (z >= Lin) z -= Lin;
    float v = src[(long)z * inner];
    lo = fmaf(c_lo[7 - k], v, lo);
    if (HI) hi = fmaf(c_hi[7 - k], v, hi);
  }
  long oo = (oa * (long)Lout + j) * inner + ii;
  outLo[oo] = lo;
  if (HI) outHi[oo] = hi;
}

// ---------------------------------------------------------------------------
// Inverse (transposed-conv + crop) pass. Upsample-by-2, 8-tap, crop [6:6+Lout).
// Crop keeps all taps in-bounds -> no wrap, exactly 4 taps per output.
// Two-level outer (A=subband-pair-major, B=dims between) with explicit strides.
// ---------------------------------------------------------------------------
template<bool HI>
__global__ __launch_bounds__(256)
void inv_pass(const float* __restrict__ inLo, const float* __restrict__ inHi,
              float* __restrict__ out,
              long nOB, long strAin, long strBin,
              long strAout, long strBout,
              int Lin, int Lout, long inner, long total)
{
  long idx = (long)blockIdx.x * blockDim.x + threadIdx.x;
  if (idx >= total) return;
  long ii = idx % inner;
  long t  = idx / inner;
  int  j  = (int)(t % Lout);
  long ot = t / Lout;
  long ob = ot % nOB;
  long oa = ot / nOB;
  (void)Lin;

  int k0 = (j & 1) ^ 1;                 // tap parity: u = j + k - 1 must be even
  int m0 = (j + k0 - 1) >> 1;           // first source index, always in-bounds

  long inoff = oa * strAin + ob * strBin + ii + (long)m0 * inner;
  const float* lo = inLo + inoff;
  float acc = 0.f;
#pragma unroll
  for (int q = 0; q < 4; ++q)
    acc = fmaf(c_lo[k0 + 2 * q], lo[(long)q * inner], acc);
  if (HI) {
    const float* hi = inHi + inoff;
#pragma unroll
    for (int q = 0; q < 4; ++q)
      acc = fmaf(c_hi[k0 + 2 * q], hi[(long)q * inner], acc);
  }
  out[oa * strAout + ob * strBout + (long)j * inner + ii] = acc;
}

// ---------------------------------------------------------------------------
// W transpose: W[s][i][o][p] -> Wt[s][p][i][o]   (p = flat 21^3 = 9261)
// LDS-tiled 32x32 so both global read (over p) and write (over o) coalesce.
// ---------------------------------------------------------------------------
__global__ __launch_bounds__(256)
void wtrans(const float* __restrict__ W, float* __restrict__ Wt)
{
  __shared__ float tile[32][33];
  int si = blockIdx.y;                  // s*32 + i, 256 total
  int p0 = blockIdx.x * 32;             // p tile
  int tx = threadIdx.x;                 // 0..31
  int ty = threadIdx.y;                 // 0..7
#pragma unroll
  for (int oi = 0; oi < 4; ++oi) {
    int o = ty + 8 * oi;
    int p = p0 + tx;
    tile[o][tx] = (p < 9261) ? W[((long)si * 32 + o) * 9261 + p] : 0.f;
  }
  __syncthreads();
  int s = si >> 5, i = si & 31;
#pragma unroll
  for (int pi = 0; pi < 4; ++pi) {
    int pl = ty + 8 * pi;
    int p  = p0 + pl;
    if (p < 9261)
      Wt[(((long)s * 9261 + p) * 32 + i) * 32 + tx] = tile[tx][pl];
  }
}

// ---------------------------------------------------------------------------
// Channel mix via V_WMMA_F32_16X16X4_F32:
//   per (s,p):  D[16x16(x2)] = A[16x4] x B[4x16] accumulated over 8 K-steps.
//   M rows 0..3  = batch b (rows 4..15 zero), K = Ci = 32, N tiles cover Co=32.
// c2t layout: [s][b*32+i][p] ; Wt layout: [s][p][i][o] ; mixed: [b*Co+o][s][p]
// One wave per (s,p); block = 8 waves; grid = 9261 blocks (8*9261 pts exactly).
//
// A-loads are unconditional from a clamped row (rb = r&3, always in-bounds) and
// zeroed by value-select afterwards -> no EXEC-mask branching around WMMA.
// USE_WT is a template param so addressing strides fold to constants.
// ---------------------------------------------------------------------------
template<bool USE_WT>
__global__ __launch_bounds__(256)
void mix_wmma(const float* __restrict__ c2t, const float* __restrict__ Wsrc,
              float* __restrict__ mixed)
{
  const int lane = threadIdx.x & 31;
  const int wv   = threadIdx.x >> 5;
  const long q   = (long)blockIdx.x * 8 + wv;     // 0 .. 74087 (exact fit)
  const int s = (int)(q / 9261);
  const int p = (int)(q % 9261);

  const int half = lane >> 4;                     // ISA A-frag: K group
  const int r    = lane & 15;                     // ISA A-frag: M row / B-frag: N col
  const bool av  = (r < 4);                       // only batch rows 0..3 valid
  const int  rb  = r & 3;                         // clamped row: safe address

  // ---- stage the 16 A values this lane contributes (one s_clause of loads)
  const float* Ab = c2t + ((long)s * 128 + (long)rb * 32) * 9261 + p;
  float areg[16];
#pragma unroll
  for (int kb = 0; kb < 8; ++kb) {
    const int i0 = 4 * kb + 2 * half;
    areg[2 * kb]     = Ab[(long)i0 * 9261];
    areg[2 * kb + 1] = Ab[(long)(i0 + 1) * 9261];
  }

  const float* Bw = USE_WT ? (Wsrc + ((long)s * 9261 + p) * 1024)    // + i*32+o
                           : (Wsrc + (long)s * 32 * 32 * 9261 + p);  // + (i*32+o)*9261
  const long st = USE_WT ? 1L : 9261L;            // element stride inside a row

  if (USE_WT)  // stream hint for the next point-block of weights (read-once data)
    __builtin_prefetch(Bw + 8 * 1024, 0, 1);

  v8f acc0 = {0.f,0.f,0.f,0.f,0.f,0.f,0.f,0.f};
  v8f acc1 = {0.f,0.f,0.f,0.f,0.f,0.f,0.f,0.f};

#pragma unroll
  for (int kb = 0; kb < 8; ++kb) {
    const int i0 = 4 * kb + 2 * half;
    v2f a;
    a.x = av ? areg[2 * kb]     : 0.f;            // value-select, EXEC untouched
    a.y = av ? areg[2 * kb + 1] : 0.f;
    v2f b0, b1;
    b0.x = Bw[(long)(i0 * 32 + r)            * st];
    b0.y = Bw[(long)((i0 + 1) * 32 + r)      * st];
    b1.x = Bw[(long)(i0 * 32 + r + 16)       * st];
    b1.y = Bw[(long)((i0 + 1) * 32 + r + 16) * st];
    acc0 = __builtin_amdgcn_wmma_f32_16x16x4_f32(false, a, false, b0,
                                                 (short)0, acc0, false, false);
    acc1 = __builtin_amdgcn_wmma_f32_16x16x4_f32(false, a, false, b1,
                                                 (short)0, acc1, false, false);
  }

  // D layout: VGPR v holds M = v + 8*half, N = r. Valid rows M = b in 0..3.
  if (half == 0) {
#pragma unroll
    for (int b = 0; b < 4; ++b) {
      mixed[(((long)b * 32 + r     ) * 8 + s) * 9261 + p] = acc0[b];
      mixed[(((long)b * 32 + r + 16) * 8 + s) * 9261 + p] = acc1[b];
    }
  }
}

// ---------------------------------------------------------------------------
static inline int nblk(long total) { return (int)((total + 255) / 256); }

extern "C" void kernel_launch(void* const* d_in, const int* in_sizes, int n_in,
                              void* d_out, int out_size, void* d_ws, size_t ws_size,
                              hipStream_t stream)
{
  (void)in_sizes; (void)n_in; (void)out_size;
  const float* x = (const float*)d_in[0];   // (4,32,64,64,64)
  const float* W = (const float*)d_in[1];   // (8,32,32,21,21,21)
  float* out = (float*)d_out;               // (4,32,64,64,64)
  float* ws  = (float*)d_ws;

  const long WT_ELEMS = 75841536L;          // 8*32*32*9261
  const long S1 = 18350080L;                // t1 / mixed / u2
  const long S2 = 10035200L;                // t2 / Bq2 / r1o / u1
  const long S3 =  9483264L;                // c1a / c2t / res
  const long S4 =  6585600L;                // Aq1 / r2o
  const long need = WT_ELEMS + S1 + S2 + S3 + S4;

  const int use_wt = (ws_size >= (size_t)need * sizeof(float)) ? 1 : 0;
  float* Wt    = ws;
  float* slot1 = ws + (use_wt ? WT_ELEMS : 0);
  float* slot2 = slot1 + S1;
  float* slot3 = slot2 + S2;
  float* slot4 = slot3 + S3;

  long tot;

  // ---- W transpose (only if workspace permits) ----
  if (use_wt)
    wtrans<<<dim3((9261 + 31) / 32, 256), dim3(32, 8), 0, stream>>>(W, Wt);

  // ---- Forward level 1 (low-pass only), 64 -> 35 per axis ----
  tot = 524288L * 35;                                     // z
  fwd_pass<false><<<nblk(tot), 256, 0, stream>>>(x, slot1, nullptr, 64, 35, 1L, tot);
  tot = 8192L * 35 * 35;                                  // y
  fwd_pass<false><<<nblk(tot), 256, 0, stream>>>(slot1, slot2, nullptr, 64, 35, 35L, tot);
  tot = 128L * 35 * 1225;                                 // x -> c1a (slot3)
  fwd_pass<false><<<nblk(tot), 256, 0, stream>>>(slot2, slot3, nullptr, 64, 35, 1225L, tot);

  // ---- Forward level 2 (lo+hi), 35 -> 21 per axis ----
  tot = 156800L * 21;                                     // z -> slot4 [sz][..]
  fwd_pass<true><<<nblk(tot), 256, 0, stream>>>(slot3, slot4, slot4 + 3292800L, 35, 21, 1L, tot);
  tot = 8960L * 21 * 21;                                  // y -> slot2 [sy][sz][..]
  fwd_pass<true><<<nblk(tot), 256, 0, stream>>>(slot4, slot2, slot2 + 3951360L, 35, 21, 21L, tot);
  tot = 512L * 21 * 441;                                  // x -> c2t (slot3) [s][vol][p]
  fwd_pass<true><<<nblk(tot), 256, 0, stream>>>(slot2, slot3, slot3 + 4741632L, 35, 21, 441L, tot);

  // ---- Channel mix (WMMA f32 16x16x4) -> mixed (slot1) [b*Co+o][s][p] ----
  if (use_wt)
    mix_wmma<true ><<<9261, 256, 0, stream>>>(slot3, Wt, slot1);
  else
    mix_wmma<false><<<9261, 256, 0, stream>>>(slot3, W,  slot1);

  // ---- Inverse level 2 (8 subbands), 21 -> 35 per axis ----
  tot = 512L * 441 * 35;                                  // z: pairs (..,z=a/d)
  inv_pass<true><<<nblk(tot), 256, 0, stream>>>(slot1, slot1 + 9261L, slot2,
      441L, 18522L, 21L, 15435L, 35L, 21, 35, 1L, tot);
  tot = 256L * 21 * 35 * 35;                              // y
  inv_pass<true><<<nblk(tot), 256, 0, stream>>>(slot2, slot2 + 15435L, slot4,
      21L, 30870L, 735L, 25725L, 1225L, 21, 35, 35L, tot);
  tot = 128L * 35 * 1225;                                 // x -> res (slot3)
  inv_pass<true><<<nblk(tot), 256, 0, stream>>>(slot4, slot4 + 25725L, slot3,
      1L, 51450L, 0L, 42875L, 0L, 21, 35, 1225L, tot);

  // ---- Inverse level 1 (low-pass only), 35 -> 64 per axis ----
  tot = 156800L * 64;                                     // z
  inv_pass<false><<<nblk(tot), 256, 0, stream>>>(slot3, nullptr, slot2,
      1L, 35L, 0L, 64L, 0L, 35, 64, 1L, tot);
  tot = 4480L * 64 * 64;                                  // y
  inv_pass<false><<<nblk(tot), 256, 0, stream>>>(slot2, nullptr, slot1,
      1L, 2240L, 0L, 4096L, 0L, 35, 64, 64L, tot);
  tot = 128L * 64 * 4096;                                 // x -> d_out
  inv_pass<false><<<nblk(tot), 256, 0, stream>>>(slot1, nullptr, out,
      1L, 143360L, 0L, 262144L, 0L, 35, 64, 4096L, tot);
}